// Model_6133213298915
// MI455X (gfx1250) — compile-verified
//
#include <hip/hip_runtime.h>

#define NODE_LEN 64
#define HIDDEN_LEN 128

typedef __attribute__((ext_vector_type(2))) float v2f;
typedef __attribute__((ext_vector_type(8))) float v8f;

// Native CDNA5 no-return f32 atomic add (fire-and-forget, STOREcnt-tracked).
// Guarantees global_atomic_add_f32 instead of a possible CAS-loop expansion.
__device__ __forceinline__ void atomic_add_f32_native(float* p, float v) {
    asm volatile("global_atomic_add_f32 %0, %1, off" :: "v"(p), "v"(v) : "memory");
}

// ---------------------------------------------------------------------------
// Phase A: one MPNN step = scatter-add of h[src] rows into hout[dst].
// 16 lanes per edge, each lane moves a float4 (64 floats per edge row).
// hout must be zero-initialized. Atomics land in L2 (h buffers are 25.6 MB).
// ---------------------------------------------------------------------------
__global__ void scatter_add_kernel(const float* __restrict__ hin,
                                   float* __restrict__ hout,
                                   const int* __restrict__ src,
                                   const int* __restrict__ dst,
                                   int n_edges) {
    long long t = (long long)blockIdx.x * blockDim.x + threadIdx.x;
    int e = (int)(t >> 4);
    if (e >= n_edges) return;
    int c4 = ((int)t & 15) << 2;

    long long s = (long long)src[e];
    long long d = (long long)dst[e];

    const float4 v = *(const float4*)(hin + s * NODE_LEN + c4);
    float* o = hout + d * NODE_LEN + c4;
    // One 64-bit address, four hardware f32 atomics with immediate offsets.
    asm volatile(
        "global_atomic_add_f32 %0, %1, off\n\t"
        "global_atomic_add_f32 %0, %2, off offset:4\n\t"
        "global_atomic_add_f32 %0, %3, off offset:8\n\t"
        "global_atomic_add_f32 %0, %4, off offset:12"
        :: "v"(o), "v"(v.x), "v"(v.y), "v"(v.z), "v"(v.w)
        : "memory");
}

// ---------------------------------------------------------------------------
// Phase B: colsum[c] = sum over nodes of h[:, c]  (c in 0..63).
// Grid stride (blocks*threads) is a multiple of 64, so each thread touches a
// single column; one atomic per thread at the end.
// ---------------------------------------------------------------------------
__global__ void colsum_kernel(const float* __restrict__ h,
                              float* __restrict__ colsum,
                              long long n_elems) {
    const long long tid = (long long)blockIdx.x * blockDim.x + threadIdx.x;
    const long long stride = (long long)gridDim.x * blockDim.x;
    float acc = 0.0f;
    for (long long i = tid; i < n_elems; i += stride) acc += h[i];
    const int col = (int)(tid & (NODE_LEN - 1));
    atomic_add_f32_native(&colsum[col], acc);
}

// ---------------------------------------------------------------------------
// Phase C: out = W_read @ (W_pool[:, :64] @ (colsum / N)) + b_read
// One wave, EXEC all-ones at every WMMA. V_WMMA_F32_16X16X4_F32, accumulate
// over K chunks.
//   A (16x4 f32): lanes 0-15 -> M=lane, K=khalf..khalf+1 with khalf=0;
//                 lanes 16-31 -> M=lane-16, khalf=2.
//   B (4x16 f32): only column N=0 nonzero -> lanes 0 and 16 carry data.
//   D column N=0: lane 0 holds M=0..7 in C[0..7], lane 16 holds M=8..15.
// ---------------------------------------------------------------------------
__global__ void final_wmma_kernel(const float* __restrict__ colsum,
                                  const float* __restrict__ W_pool,
                                  const float* __restrict__ W_read,
                                  const float* __restrict__ b_read,
                                  float* __restrict__ out,
                                  float inv_n) {
    const int lane  = threadIdx.x;          // 0..31
    const int m     = lane & 15;
    const int khalf = (lane < 16) ? 0 : 2;

    float acc = 0.0f;
    for (int mb = 0; mb < 8; ++mb) {
        v8f C = {0.f, 0.f, 0.f, 0.f, 0.f, 0.f, 0.f, 0.f};
        const int row = mb * 16 + m;        // W_pool row for this lane
        #pragma unroll
        for (int kb = 0; kb < 16; ++kb) {   // K = 64 in chunks of 4
            const int k0 = kb * 4 + khalf;
            v2f A, B;
            A[0] = W_pool[row * HIDDEN_LEN + k0];
            A[1] = W_pool[row * HIDDEN_LEN + k0 + 1];
            float b0 = 0.0f, b1 = 0.0f;
            if (m == 0) {                   // column N=0 of B
                b0 = colsum[k0] * inv_n;
                b1 = colsum[k0 + 1] * inv_n;
            }
            B[0] = b0; B[1] = b1;
            C = __builtin_amdgcn_wmma_f32_16x16x4_f32(
                    false, A, false, B, (short)0, C, false, false);
        }
        if (m == 0) {
            const int mbase = mb * 16 + ((lane < 16) ? 0 : 8);
            #pragma unroll
            for (int r = 0; r < 8; ++r) acc += C[r] * W_read[mbase + r];
        }
    }

    __shared__ float partial[32];
    partial[lane] = acc;
    __syncthreads();
    if (lane == 0) out[0] = partial[0] + partial[16] + b_read[0];
}

extern "C" void kernel_launch(void* const* d_in, const int* in_sizes, int n_in,
                              void* d_out, int out_size, void* d_ws, size_t ws_size,
                              hipStream_t stream) {
    const float* x      = (const float*)d_in[0];   // [N, 64]
    const int*   ei     = (const int*)d_in[1];     // [2, E]
    const float* W_pool = (const float*)d_in[2];   // [128, 128]
    const float* W_read = (const float*)d_in[3];   // [128]
    const float* b_read = (const float*)d_in[4];   // [1]
    float*       out    = (float*)d_out;           // [1]

    const long long n_nodes = in_sizes[0] / NODE_LEN;
    const int       n_edges = in_sizes[1] / 2;
    const int* src = ei;                 // edge_index[0]
    const int* dst = ei + n_edges;       // edge_index[1]

    const size_t h_elems = (size_t)n_nodes * NODE_LEN;
    float* hA     = (float*)d_ws;        // [N, 64]
    float* hB     = hA + h_elems;        // [N, 64]
    float* colsum = hB + h_elems;        // [64]

    const size_t h_bytes = h_elems * sizeof(float);
    const long long scatter_threads = (long long)n_edges * 16;
    const dim3 blk(256);
    const dim3 sgrid((unsigned)((scatter_threads + 255) / 256));

    // 4 MPNN steps: x -> hA -> hB -> hA -> hB  (zero target before each)
    hipMemsetAsync(hA, 0, h_bytes, stream);
    scatter_add_kernel<<<sgrid, blk, 0, stream>>>(x,  hA, src, dst, n_edges);
    hipMemsetAsync(hB, 0, h_bytes, stream);
    scatter_add_kernel<<<sgrid, blk, 0, stream>>>(hA, hB, src, dst, n_edges);
    hipMemsetAsync(hA, 0, h_bytes, stream);
    scatter_add_kernel<<<sgrid, blk, 0, stream>>>(hB, hA, src, dst, n_edges);
    hipMemsetAsync(hB, 0, h_bytes, stream);
    scatter_add_kernel<<<sgrid, blk, 0, stream>>>(hA, hB, src, dst, n_edges);

    // colsum reduction of final h
    hipMemsetAsync(colsum, 0, NODE_LEN * sizeof(float), stream);
    colsum_kernel<<<1024, 256, 0, stream>>>(hB, colsum, (long long)h_elems);

    // dense tail via f32 WMMA
    final_wmma_kernel<<<1, 32, 0, stream>>>(colsum, W_pool, W_read, b_read, out,
                                            1.0f / (float)n_nodes);
}